// FastRCNNOutputLayers_25744033972788
// MI455X (gfx1250) — compile-verified
//
#include <hip/hip_runtime.h>
#include <hip/hip_bf16.h>
#include <stdint.h>

// ---------------------------------------------------------------------------
// FastRCNN output layer post-processing for MI455X (gfx1250).
//  B=16, R=2048, K=80 (+1 bg), PRE=2048, TOPK=100.
//  Kernel 1: global top-2048 candidate selection per batch (histogram
//            threshold + bitonic sort in LDS), writes clipped boxes/score/cls
//            to workspace.
//  Kernel 2: blocked NMS per batch. Candidate boxes are DMA'd into LDS with
//            the Tensor Data Mover (tensor_load_to_lds + s_wait_tensorcnt).
//            Cross-block suppression uses V_WMMA_F32_16X16X4_F32 to form the
//            (area_i + area_j) 16x16 tile (rank-2 outer product) that feeds
//            the IoU denominator.  Class-offset trick of the reference is
//            replaced by the equivalent same-class test (MAX_COORD=1500 >
//            any coordinate extent).  The inner suppression loop is fully
//            branchless (register-resident row data + integer predicates).
// ---------------------------------------------------------------------------

#define B_BATCH 16
#define R_ROWS 2048
#define K_CLS 80
#define RK (R_ROWS * K_CLS)
#define PRE 2048
#define TOPK 100
#define BINS 4096
#define SCORE_THRESH 0.05f
#define NMS_THRESH 0.5f
#define W_IMG 1333.0f
#define H_IMG 800.0f

// per-batch workspace layout (floats): box[2048*4] | score[2048] | cls[2048]
#define WS_STRIDE 12288
#define WS_SCORE_OFF 8192
#define WS_CLS_OFF 10240

typedef float v2f __attribute__((ext_vector_type(2)));
typedef float v8f __attribute__((ext_vector_type(8)));
typedef unsigned int u32x4 __attribute__((ext_vector_type(4)));
typedef int i32x4 __attribute__((ext_vector_type(4)));
typedef int i32x8 __attribute__((ext_vector_type(8)));

__device__ __forceinline__ int imin(int a, int b) { return a < b ? a : b; }

// ===========================================================================
// Kernel 1: per-batch candidate selection + descending sort.
// ===========================================================================
__global__ __launch_bounds__(1024) void select_topk_kernel(
    const float* __restrict__ boxes,   // [B, R, K*4]
    const float* __restrict__ scores,  // [B, R, K+1]
    float* __restrict__ ws) {
  const int b = blockIdx.x;
  const int tid = threadIdx.x;

  // 16 KB of LDS shared between histogram (4096 u32) and sort keys (2048 u64)
  __shared__ unsigned long long keys[PRE];
  unsigned int* hist = reinterpret_cast<unsigned int*>(keys);
  __shared__ int sT;
  __shared__ unsigned int sAbove, sHigh, sTie;

  for (int i = tid; i < BINS; i += 1024) hist[i] = 0u;
  if (tid == 0) { sT = -1; sAbove = 0u; sHigh = 0u; sTie = 0u; }
  __syncthreads();

  // Pass 1: histogram of thresholded scores (linear bins over (0,1]).
  const float* sc_base = scores + (size_t)b * R_ROWS * (K_CLS + 1);
  for (int t = tid; t < RK; t += 1024) {
    const int r = t / K_CLS;
    const int c = t - r * K_CLS;
    const float s = sc_base[r * (K_CLS + 1) + c];
    if (s > SCORE_THRESH) {
      const int bin = imin(BINS - 1, (int)(s * (float)BINS));
      atomicAdd(&hist[bin], 1u);
    }
  }
  __syncthreads();

  // Threshold bin: largest bin with suffix count >= PRE.
  if (tid == 0) {
    unsigned int acc = 0u;
    for (int bin = BINS - 1; bin >= 0; --bin) {
      const unsigned int h = hist[bin];
      if (acc + h >= (unsigned)PRE) { sT = bin; sAbove = acc; break; }
      acc += h;
    }
  }
  __syncthreads();
  const int T = sT;
  const unsigned int above = sAbove;
  __syncthreads();

  // Reuse histogram LDS as sort keys.
  for (int i = tid; i < PRE; i += 1024) keys[i] = 0ull;
  __syncthreads();

  // Pass 2: gather candidates into slots; key = (score_bits << 32) | ~idx
  // so descending key order == (score desc, index asc) like lax.top_k.
  for (int t = tid; t < RK; t += 1024) {
    const int r = t / K_CLS;
    const int c = t - r * K_CLS;
    const float s = sc_base[r * (K_CLS + 1) + c];
    if (s > SCORE_THRESH) {
      const int bin = imin(BINS - 1, (int)(s * (float)BINS));
      unsigned int slot = 0xFFFFFFFFu;
      if (bin > T) {
        slot = atomicAdd(&sHigh, 1u);  // guaranteed < above <= PRE
      } else if (bin == T) {
        slot = above + atomicAdd(&sTie, 1u);
      }
      if (slot < (unsigned)PRE) {
        keys[slot] = ((unsigned long long)__float_as_uint(s) << 32) |
                     (unsigned long long)(0xFFFFFFFFu - (unsigned)t);
      }
    }
  }
  __syncthreads();

  // Bitonic sort, descending, 2048 u64 keys, 1024 threads (one pair each).
  for (unsigned int k = 2; k <= (unsigned)PRE; k <<= 1) {
    for (unsigned int j = k >> 1; j > 0; j >>= 1) {
      __syncthreads();
      const unsigned int t = tid;
      const unsigned int i = ((t & ~(j - 1u)) << 1) | (t & (j - 1u));
      const unsigned int ixj = i | j;
      const unsigned long long a = keys[i];
      const unsigned long long bb = keys[ixj];
      const bool up = ((i & k) == 0u);
      if (up ? (a < bb) : (a > bb)) { keys[i] = bb; keys[ixj] = a; }
    }
  }
  __syncthreads();

  // Write sorted candidates (clipped boxes, score, class) to workspace.
  float* cbox = ws + (size_t)b * WS_STRIDE;
  float* cscore = cbox + WS_SCORE_OFF;
  int* ccls = reinterpret_cast<int*>(cbox + WS_CLS_OFF);
  const float* box_base = boxes + (size_t)b * R_ROWS * (K_CLS * 4);
  for (int i = tid; i < PRE; i += 1024) {
    const unsigned long long key = keys[i];
    if (key != 0ull) {
      const unsigned int idx = 0xFFFFFFFFu - (unsigned int)(key & 0xFFFFFFFFull);
      const unsigned int r = idx / (unsigned)K_CLS;
      const unsigned int c = idx - r * (unsigned)K_CLS;
      const float* bp = box_base + (size_t)r * (K_CLS * 4) + c * 4;
      cbox[i * 4 + 0] = fminf(fmaxf(bp[0], 0.f), W_IMG);
      cbox[i * 4 + 1] = fminf(fmaxf(bp[1], 0.f), H_IMG);
      cbox[i * 4 + 2] = fminf(fmaxf(bp[2], 0.f), W_IMG);
      cbox[i * 4 + 3] = fminf(fmaxf(bp[3], 0.f), H_IMG);
      cscore[i] = __uint_as_float((unsigned int)(key >> 32));
      ccls[i] = (int)c;
    } else {
      cbox[i * 4 + 0] = 0.f; cbox[i * 4 + 1] = 0.f;
      cbox[i * 4 + 2] = 0.f; cbox[i * 4 + 3] = 0.f;
      cscore[i] = -__builtin_inff();
      ccls[i] = -1;
    }
  }
}

// ===========================================================================
// Kernel 2: blocked NMS + top-100 emission. 256 threads (8 wave32) per batch.
// ===========================================================================
__global__ __launch_bounds__(256) void nms_kernel(
    const float* __restrict__ ws, float* __restrict__ out) {
  const int b = blockIdx.x;
  const int tid = threadIdx.x;

  __shared__ float sbox[PRE * 4];          // 32 KB, filled by TDM
  __shared__ float sarea[PRE];             // 8 KB
  __shared__ int scls[PRE];                // 8 KB (255 == invalid)
  __shared__ int ssup[PRE];                // 8 KB
  __shared__ int skept[PRE];               // 8 KB
  __shared__ int winners[TOPK];
  __shared__ int wcount;

  const float* cbox = ws + (size_t)b * WS_STRIDE;
  const float* cscore = cbox + WS_SCORE_OFF;
  const int* ccls = reinterpret_cast<const int*>(cbox + WS_CLS_OFF);

  __builtin_prefetch(cscore + tid, 0, 1);  // -> global_prefetch_b8

  for (int i = tid; i < PRE; i += 256) {
    const int cv = ccls[i];
    scls[i] = (cv >= 0) ? cv : 255;
    ssup[i] = 0;
    skept[i] = 0;
  }
  if (tid == 0) wcount = 0;

  // --- Tensor Data Mover: DMA the 32KB candidate-box tile into LDS. ---
  // D# per CDNA5 ISA ch.8: 1-D tile, 8192 f32 elements, contiguous.
  if (tid == 0) {
    const unsigned long long ga = (unsigned long long)(uintptr_t)cbox;
    const unsigned int lds_addr = (unsigned int)(uintptr_t)(&sbox[0]);
    u32x4 g0;
    g0.x = 1u;                                            // count=1
    g0.y = lds_addr;                                      // lds_addr
    g0.z = (unsigned int)(ga & 0xFFFFFFFFull);            // global_addr lo
    g0.w = (unsigned int)((ga >> 32) & 0x01FFFFFFull) | (2u << 30);  // hi|type=2
    i32x8 g1;
    g1[0] = (int)(2u << 16);   // data_size=4B, wg_mask=0, no pad/iterate
    g1[1] = (int)0x20000000;   // tensor_dim0 = 8192 (low 16 in [31:16])
    g1[2] = (int)0x00010000;   // tensor_dim0 hi=0, tensor_dim1 = 1
    g1[3] = (int)0x20000000;   // tensor_dim1 hi=0, tile_dim0 = 8192
    g1[4] = 1;                 // tile_dim1 = 1, tile_dim2 = 0
    g1[5] = 8192;              // tensor_dim0_stride = 8192
    g1[6] = 0;
    g1[7] = 0;
    i32x4 g2 = {0, 0, 0, 0};
    i32x4 g3 = {0, 0, 0, 0};
    i32x8 g4 = {0, 0, 0, 0, 0, 0, 0, 0};
    __builtin_amdgcn_tensor_load_to_lds(g0, g1, g2, g3, g4, 0);
    __builtin_amdgcn_s_wait_tensorcnt(0);
  }
  __syncthreads();

  for (int i = tid; i < PRE; i += 256) {
    const float w = fmaxf(sbox[i * 4 + 2] - sbox[i * 4 + 0], 0.f);
    const float h = fmaxf(sbox[i * 4 + 3] - sbox[i * 4 + 1], 0.f);
    sarea[i] = w * h;
  }
  __syncthreads();

  // --- Blocked greedy NMS: 128 diagonal blocks of 16 rows. ---
  const int wid = tid >> 5;
  const int lane = tid & 31;
  for (int bi = 0; bi < PRE / 16; ++bi) {
    const int m0 = bi * 16;
    // Serial resolution inside the 16x16 diagonal block.
    if (tid == 0) {
      for (int p = 0; p < 16; ++p) {
        const int gi = m0 + p;
        if (scls[gi] != 255 && !ssup[gi]) {
          skept[gi] = 1;
          const float ax1 = sbox[gi * 4 + 0], ay1 = sbox[gi * 4 + 1];
          const float ax2 = sbox[gi * 4 + 2], ay2 = sbox[gi * 4 + 3];
          for (int q = p + 1; q < 16; ++q) {
            const int gj = m0 + q;
            if (scls[gj] == scls[gi]) {
              const float iw = fminf(ax2, sbox[gj * 4 + 2]) - fmaxf(ax1, sbox[gj * 4 + 0]);
              const float ih = fminf(ay2, sbox[gj * 4 + 3]) - fmaxf(ay1, sbox[gj * 4 + 1]);
              const float inter = fmaxf(iw, 0.f) * fmaxf(ih, 0.f);
              const float uni = sarea[gi] + sarea[gj] - inter + 1e-9f;
              if (inter > NMS_THRESH * uni) ssup[gj] = 1;
            }
          }
        }
      }
    }
    __syncthreads();

    // Preload this block's 8 rows per lane into registers (invariant for all
    // column tiles of this block): lanes 0-15 rows m0..m0+7, lanes 16-31 rows
    // m0+8..m0+15 (matches the WMMA C-layout M mapping).
    const int mbase = m0 + ((lane & 16) ? 8 : 0);
    float mx1[8], my1[8], mx2[8], my2[8];
    float areaA;
    int mok[8];  // kept-and-valid predicate per row
    {
#pragma unroll
      for (int r = 0; r < 8; ++r) {
        const int m = mbase + r;
        mx1[r] = sbox[m * 4 + 0];
        my1[r] = sbox[m * 4 + 1];
        mx2[r] = sbox[m * 4 + 2];
        my2[r] = sbox[m * 4 + 3];
        mok[r] = (skept[m] != 0) ? (scls[m] + 1) : 0;  // 0 = skip, else cls+1
      }
      areaA = sarea[m0 + (lane & 15)];
    }

    // Parallel suppression of all later columns, 16-wide tiles per wave.
    const int jstart = m0 + 16;
    const int ntiles = (PRE - jstart) >> 4;
    for (int tt = wid; tt < ntiles; tt += 8) {
      const int n0 = jstart + tt * 16;
      // WMMA rank-2 outer product: C[m][n] = area_m * 1 + 1 * area_n.
      // A (16x4 f32): lanes 0-15 hold row m=lane, K0/K1 = {area_m, 1}.
      // B (4x16 f32): vgpr0 lanes 0-15 = row K0 (=1), lanes 16-31 = row K1
      //               (=area_n); vgpr1 = rows K2/K3 (=0).
      v2f af, bf;
      af.x = (lane < 16) ? areaA : 0.f;
      af.y = (lane < 16) ? 1.f : 0.f;
      bf.x = (lane < 16) ? 1.f : sarea[n0 + (lane - 16)];
      bf.y = 0.f;
      v8f acc = {};
      acc = __builtin_amdgcn_wmma_f32_16x16x4_f32(
          false, af, false, bf, (short)0, acc, false, false);
      // C layout: lane l, vgpr r -> M = r + 8*(l>=16), N = l & 15.
      const int n = n0 + (lane & 15);
      const int cn1 = scls[n] + 1;  // 256 == invalid (never matches mok)
      const float nx1 = sbox[n * 4 + 0], ny1 = sbox[n * 4 + 1];
      const float nx2 = sbox[n * 4 + 2], ny2 = sbox[n * 4 + 3];
      int supflag = 0;
#pragma unroll
      for (int r = 0; r < 8; ++r) {
        const float iw = fminf(mx2[r], nx2) - fmaxf(mx1[r], nx1);
        const float ih = fminf(my2[r], ny2) - fmaxf(my1[r], ny1);
        const float inter = fmaxf(iw, 0.f) * fmaxf(ih, 0.f);
        const float uni = acc[r] - inter + 1e-9f;  // area_m + area_n - inter
        const int hit = (inter > NMS_THRESH * uni) ? 1 : 0;
        const int ok = (mok[r] == cn1) ? 1 : 0;    // kept && same class
        supflag |= (hit & ok);
      }
      if (supflag && cn1 != 256) ssup[n] = 1;
    }
    __syncthreads();
  }

  // --- Compact first 100 kept candidates (already score-descending). ---
  if (tid == 0) {
    int cnt = 0;
    for (int i = 0; i < PRE && cnt < TOPK; ++i)
      if (skept[i]) winners[cnt++] = i;
    wcount = cnt;
    for (int i = cnt; i < TOPK; ++i) winners[i] = -1;
  }
  __syncthreads();

  // Output: boxes [16,100,4] | scores [16,100] | cls(int) [16,100]
  float* out_boxes = out;
  float* out_scores = out + B_BATCH * TOPK * 4;
  int* out_cls = reinterpret_cast<int*>(out) + B_BATCH * TOPK * 5;
  if (tid < TOPK) {
    const int w = winners[tid];
    const int o = b * TOPK + tid;
    if (w >= 0) {
      out_boxes[o * 4 + 0] = sbox[w * 4 + 0];
      out_boxes[o * 4 + 1] = sbox[w * 4 + 1];
      out_boxes[o * 4 + 2] = sbox[w * 4 + 2];
      out_boxes[o * 4 + 3] = sbox[w * 4 + 3];
      out_scores[o] = cscore[w];
      out_cls[o] = scls[w];
    } else {
      out_boxes[o * 4 + 0] = 0.f; out_boxes[o * 4 + 1] = 0.f;
      out_boxes[o * 4 + 2] = 0.f; out_boxes[o * 4 + 3] = 0.f;
      out_scores[o] = 0.f;
      out_cls[o] = -1;
    }
  }
}

extern "C" void kernel_launch(void* const* d_in, const int* in_sizes, int n_in,
                              void* d_out, int out_size, void* d_ws, size_t ws_size,
                              hipStream_t stream) {
  const float* boxes = (const float*)d_in[0];   // [16, 2048, 320]
  const float* scores = (const float*)d_in[1];  // [16, 2048, 81]
  float* ws = (float*)d_ws;                     // needs 16*12288*4 = 768 KB
  float* out = (float*)d_out;                   // 6400 + 1600 + 1600 elems

  select_topk_kernel<<<B_BATCH, 1024, 0, stream>>>(boxes, scores, ws);
  nms_kernel<<<B_BATCH, 256, 0, stream>>>(ws, out);
}